// DistributionSimilarity_40776419508652
// MI455X (gfx1250) — compile-verified
//
#include <hip/hip_runtime.h>
#include <stdint.h>

#define S_SAMPLES 1024
#define F_SUPPORT 64
#define IT 4            // i-rows per block
#define CJ 64           // j-rows per TDM chunk
#define NCHUNK (S_SAMPLES / CJ)
#define ROWP 68         // padded LDS row stride in floats (64 data + 4 pad)

typedef unsigned int u32;
typedef uint32_t tdm_v4u __attribute__((ext_vector_type(4)));
typedef int      tdm_v4i __attribute__((ext_vector_type(4)));
typedef int      tdm_v8i __attribute__((ext_vector_type(8)));

#if defined(__has_builtin)
#if __has_builtin(__builtin_amdgcn_tensor_load_to_lds)
#define HAVE_TDM 1
#endif
#endif
#ifndef HAVE_TDM
#define HAVE_TDM 0
#endif

#if HAVE_TDM
// Issue one TDM descriptor: copy CJ rows x 64 f32 (row-major, contiguous in
// global) into LDS at lds_byte_addr, inserting 4 DWORDs of padding after every
// 64 DWORDs (row stride in LDS becomes ROWP=68 floats -> bank-conflict-light).
__device__ __forceinline__ void tdm_load_chunk(u32 lds_byte_addr,
                                               const float* gsrc,
                                               u32 rows_remaining) {
  uint64_t ga = (uint64_t)(uintptr_t)gsrc;
  tdm_v4u g0;
  g0.x = 1u;                                   // count=1 (valid), user mode
  g0.y = lds_byte_addr;                        // lds_addr (bytes)
  g0.z = (u32)(ga & 0xffffffffu);              // global_addr[31:0]
  g0.w = (u32)((ga >> 32) & 0x01ffffffu)       // global_addr[56:32]
         | (2u << 30);                         // type = 2 ("image")
  tdm_v8i g1;
  g1[0] = (int)((2u << 16)                     // data_size = 4 bytes
                | (1u << 20)                   // pad_enable
                | (5u << 22)                   // pad_interval: every 64 DWORDs
                | (3u << 25));                 // pad_amount: 4 DWORDs
  g1[1] = (int)((u32)F_SUPPORT << 16);         // tensor_dim0 (low16) = 64
  g1[2] = (int)(rows_remaining << 16);         // tensor_dim0 hi=0 | tensor_dim1 lo16
  g1[3] = (int)((u32)F_SUPPORT << 16);         // tensor_dim1 hi=0 | tile_dim0 = 64
  g1[4] = (int)CJ;                             // tile_dim1 = 64, tile_dim2 = 0
  g1[5] = (int)F_SUPPORT;                      // tensor_dim0_stride (lo32) = 64 elems
  g1[6] = 0;                                   // stride0 hi | stride1 lo
  g1[7] = 0;                                   // stride1 hi
  tdm_v4i gz = {0, 0, 0, 0};
#if __clang_major__ >= 23
  tdm_v8i gz8 = {0, 0, 0, 0, 0, 0, 0, 0};
  __builtin_amdgcn_tensor_load_to_lds(g0, g1, gz, gz, gz8, 0);
#else
  __builtin_amdgcn_tensor_load_to_lds(g0, g1, gz, gz, 0);
#endif
}
#endif

// ed[q,i,j] = (1/64) * sum_k exp(-(v[q,i,k]-v[q,j,k])^2)
__global__ void __launch_bounds__(256)
ed_kernel(const float* __restrict__ vd, float* __restrict__ ed) {
  __shared__ float s_vi[IT * F_SUPPORT];
  __shared__ float s_vj[2][CJ * ROWP];

  const int tid = threadIdx.x;
  const int q  = blockIdx.x / (S_SAMPLES / IT);
  const int i0 = (blockIdx.x % (S_SAMPLES / IT)) * IT;
  const float* vq = vd + (size_t)q * S_SAMPLES * F_SUPPORT;

  // Stage the 4 i-rows (256 floats) into LDS: one element per thread.
  s_vi[tid] = vq[(size_t)i0 * F_SUPPORT + tid];

#if HAVE_TDM
  if (tid < 32) {  // wave 0 only: TDM is wave-level (EXEC ignored)
    tdm_load_chunk((u32)(uintptr_t)&s_vj[0][0], vq, S_SAMPLES);
  }
#endif

  const int jl = tid & (CJ - 1);   // j within chunk
  const int il = tid >> 6;         // local i row (uniform per wave32)

  for (int c = 0; c < NCHUNK; ++c) {
    const int buf = c & 1;
#if HAVE_TDM
    if (tid < 32) {
      if (c + 1 < NCHUNK) {
        tdm_load_chunk((u32)(uintptr_t)&s_vj[buf ^ 1][0],
                       vq + (size_t)(c + 1) * CJ * F_SUPPORT,
                       (u32)(S_SAMPLES - (c + 1) * CJ));
        __builtin_amdgcn_s_wait_tensorcnt(1);  // chunk c complete (in-order)
      } else {
        __builtin_amdgcn_s_wait_tensorcnt(0);
      }
    }
    __syncthreads();
#else
    for (int e = tid; e < CJ * F_SUPPORT; e += 256) {
      int r = e >> 6, k = e & 63;
      s_vj[buf][r * ROWP + k] = vq[(size_t)(c * CJ + r) * F_SUPPORT + k];
    }
    __syncthreads();
#endif

    const float4* vi4 = reinterpret_cast<const float4*>(&s_vi[il * F_SUPPORT]);
    const float4* vj4 = reinterpret_cast<const float4*>(&s_vj[buf][jl * ROWP]);
    float acc = 0.0f;
#pragma unroll
    for (int k4 = 0; k4 < F_SUPPORT / 4; ++k4) {
      float4 a = vi4[k4];
      float4 b = vj4[k4];
      float d0 = a.x - b.x, d1 = a.y - b.y;
      float d2 = a.z - b.z, d3 = a.w - b.w;
      acc += __expf(-(d0 * d0));   // v_exp_f32 (TRANS, co-executes with VALU)
      acc += __expf(-(d1 * d1));
      acc += __expf(-(d2 * d2));
      acc += __expf(-(d3 * d3));
    }
    const int j = c * CJ + jl;
    ed[((size_t)q * S_SAMPLES + (i0 + il)) * S_SAMPLES + j] =
        acc * (1.0f / F_SUPPORT);
    __syncthreads();  // all reads of s_vj[buf] done before it is overwritten
  }
}

// ed_later[row, j] = softmax_j(ed[row, :]) * (j != i)
__global__ void __launch_bounds__(256)
softmax_kernel(const float* __restrict__ ed, float* __restrict__ out) {
  __shared__ float s_red[256];
  const int tid = threadIdx.x;
  const int row = blockIdx.x;              // q * S + i
  const int i = row & (S_SAMPLES - 1);
  const float* src = ed + (size_t)row * S_SAMPLES;

  float v[4];
  float m = -3.0e38f;
#pragma unroll
  for (int t = 0; t < 4; ++t) {
    v[t] = src[tid + t * 256];
    m = fmaxf(m, v[t]);
  }
  s_red[tid] = m;
  __syncthreads();
  for (int s = 128; s > 0; s >>= 1) {
    if (tid < s) s_red[tid] = fmaxf(s_red[tid], s_red[tid + s]);
    __syncthreads();
  }
  m = s_red[0];
  __syncthreads();

  float e[4];
  float sum = 0.0f;
#pragma unroll
  for (int t = 0; t < 4; ++t) {
    e[t] = __expf(v[t] - m);
    sum += e[t];
  }
  s_red[tid] = sum;
  __syncthreads();
  for (int s = 128; s > 0; s >>= 1) {
    if (tid < s) s_red[tid] += s_red[tid + s];
    __syncthreads();
  }
  const float inv = 1.0f / s_red[0];

  float* dst = out + (size_t)row * S_SAMPLES;
#pragma unroll
  for (int t = 0; t < 4; ++t) {
    const int j = tid + t * 256;
    dst[j] = (j == i) ? 0.0f : e[t] * inv;
  }
}

extern "C" void kernel_launch(void* const* d_in, const int* in_sizes, int n_in,
                              void* d_out, int out_size, void* d_ws, size_t ws_size,
                              hipStream_t stream) {
  (void)n_in; (void)d_ws; (void)ws_size; (void)out_size;
  const float* vd = (const float*)d_in[0];  // [nq, 1024, 64] f32
  const int nq = in_sizes[0] / (S_SAMPLES * F_SUPPORT);

  float* ed = (float*)d_out;                               // output 0
  float* ed_later = ed + (size_t)nq * S_SAMPLES * S_SAMPLES;  // output 1

  ed_kernel<<<dim3(nq * (S_SAMPLES / IT)), dim3(256), 0, stream>>>(vd, ed);
  softmax_kernel<<<dim3(nq * S_SAMPLES), dim3(256), 0, stream>>>(ed, ed_later);
}